// SentiAttention_76450417868986
// MI455X (gfx1250) — compile-verified
//
#include <hip/hip_runtime.h>

// Problem sizes (fixed by the reference)
#define B_    64
#define N_    2048
#define A_    512
#define RNN_  1024
#define DW_   1024
#define DS_   1024

#define NCHUNKS 8
#define CHUNK   (N_ / NCHUNKS)   // 256 rows of x per pooling block

typedef __attribute__((ext_vector_type(2))) float v2f;
typedef __attribute__((ext_vector_type(4))) float v4f;
typedef __attribute__((ext_vector_type(8))) float v8f;

#if defined(__has_builtin)
#if __has_builtin(__builtin_amdgcn_tanhf)
#define TANHF(x) __builtin_amdgcn_tanhf(x)
#else
#define TANHF(x) tanhf(x)
#endif
#else
#define TANHF(x) tanhf(x)
#endif

// ---------------------------------------------------------------------------
// K1: h_word[B,A] = h[B,RNN] @ W[A,RNN]^T + bias[A]
// One wave per 16x16 output tile using V_WMMA_F32_16X16X4_F32.
// Lane layouts per CDNA5 ISA 7.12.2:
//   A (16x4 f32):  lane L holds row m=L&15; VGPR i holds K = 2*(L>>4)+i
//   B (4x16 f32):  lane L holds col n=L&15; VGPR i holds K = 2*(L>>4)+i
//   C/D (16x16):   VGPR r, lane L -> M = r + 8*(L>>4), N = L&15
// B(k,n) = W[n,k], so the B operand reads W rows directly (W is [A,RNN]).
// ---------------------------------------------------------------------------
__global__ __launch_bounds__(32)
void hword_wmma_kernel(const float* __restrict__ h,     // [B, RNN]
                       const float* __restrict__ W,     // [A, RNN]
                       const float* __restrict__ bias,  // [A]
                       float* __restrict__ h_word)      // [B, A]
{
    const int m0   = blockIdx.x * 16;   // tile along B
    const int n0   = blockIdx.y * 16;   // tile along A
    const int lane = threadIdx.x;       // wave32
    const int mn   = lane & 15;
    const int hv   = lane >> 4;         // half-wave selector

    const float* __restrict__ arow = h + (size_t)(m0 + mn) * RNN_;
    const float* __restrict__ brow = W + (size_t)(n0 + mn) * RNN_;

    v8f acc = {};
    for (int k = 0; k < RNN_; k += 4) {
        const int kk = k + 2 * hv;
        v2f a; a.x = arow[kk]; a.y = arow[kk + 1];
        v2f b; b.x = brow[kk]; b.y = brow[kk + 1];
        acc = __builtin_amdgcn_wmma_f32_16x16x4_f32(
            /*neg_a=*/false, a, /*neg_b=*/false, b,
            /*c_mod=*/(short)0, acc, /*reuse_a=*/false, /*reuse_b=*/false);
    }

    const float bv = bias[n0 + mn];
#pragma unroll
    for (int r = 0; r < 8; ++r) {
        const int m = m0 + r + 8 * hv;
        h_word[(size_t)m * A_ + n0 + mn] = acc[r] + bv;
    }
}

// ---------------------------------------------------------------------------
// K2: scores[b,n] = sum_a tanh(p[b,n,a] + h_word[b,a]) * w_alpha[a] + b_alpha
// One wave per n-row (float4 coalesced loads of p: 256 MB single pass).
// h_word row + w_alpha staged in LDS once per block.
// ---------------------------------------------------------------------------
__global__ __launch_bounds__(256)
void scores_kernel(const float* __restrict__ p,        // [B, N, A]
                   const float* __restrict__ h_word,   // [B, A]
                   const float* __restrict__ w_alpha,  // [A]
                   const float* __restrict__ b_alpha,  // scalar
                   float* __restrict__ scores)         // [B, N]
{
    __shared__ float s_hw[A_];
    __shared__ float s_wa[A_];

    const int b   = blockIdx.x;
    const int tid = threadIdx.x;
    for (int i = tid; i < A_; i += 256) {
        s_hw[i] = h_word[(size_t)b * A_ + i];
        s_wa[i] = w_alpha[i];
    }
    __syncthreads();

    const float balpha = b_alpha[0];
    const int wave   = tid >> 5;
    const int lane   = tid & 31;
    const int n_base = blockIdx.y * 64;   // 64 rows per block, 8 per wave

#pragma unroll
    for (int r = 0; r < 8; ++r) {
        const int n = n_base + wave + 8 * r;
        const float* __restrict__ row = p + ((size_t)b * N_ + n) * A_;
        float s = 0.f;
#pragma unroll
        for (int j = 0; j < 4; ++j) {
            const int a0 = lane * 4 + j * 128;        // 32 lanes * float4 * 4 = 512
            v4f pv = *(const v4f*)(row + a0);
            s += TANHF(pv.x + s_hw[a0 + 0]) * s_wa[a0 + 0];
            s += TANHF(pv.y + s_hw[a0 + 1]) * s_wa[a0 + 1];
            s += TANHF(pv.z + s_hw[a0 + 2]) * s_wa[a0 + 2];
            s += TANHF(pv.w + s_hw[a0 + 3]) * s_wa[a0 + 3];
        }
        // wave32 butterfly reduction
#pragma unroll
        for (int off = 16; off > 0; off >>= 1)
            s += __shfl_xor(s, off, 32);
        if (lane == 0)
            scores[(size_t)b * N_ + n] = s + balpha;
    }
}

// ---------------------------------------------------------------------------
// K3: softmax over N=2048, in-place on scores -> weights. One block per b.
// ---------------------------------------------------------------------------
__global__ __launch_bounds__(256)
void softmax_kernel(float* __restrict__ scores)        // [B, N]
{
    __shared__ float red[256];
    const int b   = blockIdx.x;
    const int tid = threadIdx.x;
    float* __restrict__ row = scores + (size_t)b * N_;

    float v[8];
    float m = -INFINITY;
#pragma unroll
    for (int i = 0; i < 8; ++i) {
        v[i] = row[tid + 256 * i];
        m = fmaxf(m, v[i]);
    }
    red[tid] = m;
    __syncthreads();
    for (int s = 128; s > 0; s >>= 1) {
        if (tid < s) red[tid] = fmaxf(red[tid], red[tid + s]);
        __syncthreads();
    }
    m = red[0];
    __syncthreads();

    float sum = 0.f;
#pragma unroll
    for (int i = 0; i < 8; ++i) {
        v[i] = __expf(v[i] - m);
        sum += v[i];
    }
    red[tid] = sum;
    __syncthreads();
    for (int s = 128; s > 0; s >>= 1) {
        if (tid < s) red[tid] += red[tid + s];
        __syncthreads();
    }
    const float inv = 1.0f / red[0];
#pragma unroll
    for (int i = 0; i < 8; ++i)
        row[tid + 256 * i] = v[i] * inv;
}

// ---------------------------------------------------------------------------
// K4: partial[b,c,d] = sum over N-chunk c of weight[b,n] * x[b,n,d]
// 512 blocks stream the 512 MB tensor; each thread owns 4 consecutive d
// (float4, b128 coalesced). Weight chunk staged in LDS. Deterministic
// (no atomics); partials combined in K5.
// ---------------------------------------------------------------------------
__global__ __launch_bounds__(256)
void pool_partial_kernel(const float* __restrict__ x,       // [B, N, DW]
                         const float* __restrict__ weight,  // [B, N]
                         float* __restrict__ partial)       // [B, NCHUNKS, DW]
{
    __shared__ float s_w[CHUNK];
    const int b   = blockIdx.x;
    const int c   = blockIdx.y;
    const int tid = threadIdx.x;
    const int n0  = c * CHUNK;

    s_w[tid] = weight[(size_t)b * N_ + n0 + tid];
    __syncthreads();

    const int d0 = tid * 4;
    const float* __restrict__ base = x + ((size_t)b * N_ + n0) * DW_ + d0;

    v4f acc = {0.f, 0.f, 0.f, 0.f};
    for (int n = 0; n < CHUNK; ++n) {
        if (n + 8 < CHUNK)
            __builtin_prefetch(base + (size_t)(n + 8) * DW_, 0, 0); // global_prefetch_b8
        v4f xv = *(const v4f*)(base + (size_t)n * DW_);
        const float w = s_w[n];
        acc.x = fmaf(w, xv.x, acc.x);
        acc.y = fmaf(w, xv.y, acc.y);
        acc.z = fmaf(w, xv.z, acc.z);
        acc.w = fmaf(w, xv.w, acc.w);
    }
    *(v4f*)(partial + ((size_t)b * NCHUNKS + c) * DW_ + d0) = acc;
}

// ---------------------------------------------------------------------------
// K5: out[b, 0:DS] = senti_feats[b,:]; out[b, DS+d] = sum_c partial[b,c,d]
// ---------------------------------------------------------------------------
__global__ __launch_bounds__(256)
void finalize_kernel(const float* __restrict__ senti_feats,  // [B, DS]
                     const float* __restrict__ partial,      // [B, NCHUNKS, DW]
                     float* __restrict__ out)                // [B, DS+DW]
{
    const int b   = blockIdx.x;
    const int tid = threadIdx.x;
    const int d0  = tid * 4;

    // concat part 1: copy senti_feats
    v4f sv = *(const v4f*)(senti_feats + (size_t)b * DS_ + d0);
    *(v4f*)(out + (size_t)b * (DS_ + DW_) + d0) = sv;

    // concat part 2: reduce partials
    v4f acc = {0.f, 0.f, 0.f, 0.f};
#pragma unroll
    for (int c = 0; c < NCHUNKS; ++c) {
        v4f pv = *(const v4f*)(partial + ((size_t)b * NCHUNKS + c) * DW_ + d0);
        acc.x += pv.x; acc.y += pv.y; acc.z += pv.z; acc.w += pv.w;
    }
    *(v4f*)(out + (size_t)b * (DS_ + DW_) + DS_ + d0) = acc;
}

// ---------------------------------------------------------------------------
extern "C" void kernel_launch(void* const* d_in, const int* in_sizes, int n_in,
                              void* d_out, int out_size, void* d_ws, size_t ws_size,
                              hipStream_t stream)
{
    const float* h   = (const float*)d_in[0];  // [B, RNN]
    const float* sf  = (const float*)d_in[1];  // [B, DS]
    const float* swf = (const float*)d_in[2];  // [B, N, DW]
    const float* p   = (const float*)d_in[3];  // [B, N, A]
    const float* W   = (const float*)d_in[4];  // [A, RNN]
    const float* bh  = (const float*)d_in[5];  // [A]
    const float* wa  = (const float*)d_in[6];  // [A]
    const float* ba  = (const float*)d_in[7];  // scalar

    float* out = (float*)d_out;

    // workspace layout (floats): h_word | scores/weights | partials  (~2.75 MB)
    float* ws      = (float*)d_ws;
    float* h_word  = ws;                       // B*A      = 32768
    float* scores  = h_word + B_ * A_;         // B*N      = 131072
    float* partial = scores + B_ * N_;         // B*8*DW   = 524288

    hipLaunchKernelGGL(hword_wmma_kernel, dim3(B_ / 16, A_ / 16), dim3(32), 0, stream,
                       h, W, bh, h_word);
    hipLaunchKernelGGL(scores_kernel, dim3(B_, N_ / 64), dim3(256), 0, stream,
                       p, h_word, wa, ba, scores);
    hipLaunchKernelGGL(softmax_kernel, dim3(B_), dim3(256), 0, stream,
                       scores);
    hipLaunchKernelGGL(pool_partial_kernel, dim3(B_, NCHUNKS), dim3(256), 0, stream,
                       swf, scores, partial);
    hipLaunchKernelGGL(finalize_kernel, dim3(B_), dim3(256), 0, stream,
                       sf, partial, out);
}